// GCNEncoder_46875273068979
// MI455X (gfx1250) — compile-verified
//
#include <hip/hip_runtime.h>
#include <hip/hip_bf16.h>

typedef float v2f __attribute__((ext_vector_type(2)));
typedef float v8f __attribute__((ext_vector_type(8)));

#define GB 8          // batch
#define GN 10000      // nodes
#define GD 64         // feature dim
#define GE 160000     // edges (without self loops)
#define GROWS (GB * GN)   // 80000 rows of the flattened (B*N, D) matrix

// Single-instruction ReLU: v_max_num_f32 dst, src, 0  (IEEE maximumNumber,
// same as jnp.maximum(x, 0)); avoids clang's NaN-quieting max(x,x) preamble.
__device__ __forceinline__ float relu1(float x) {
    float r;
    asm("v_max_num_f32 %0, %1, 0" : "=v"(r) : "v"(x));
    return r;
}

// ---------------------------------------------------------------------------
// Degree / normalization kernels
// ---------------------------------------------------------------------------
__global__ __launch_bounds__(256) void k_deg_init(float* deg) {
    int i = blockIdx.x * 256 + threadIdx.x;
    if (i < GN) deg[i] = 1.0f;               // self-loop contributes 1
}

__global__ __launch_bounds__(256) void k_deg_count(const int* __restrict__ dst,
                                                   float* __restrict__ deg) {
    int e = blockIdx.x * 256 + threadIdx.x;  // grid sized exactly E/256
    atomicAdd(&deg[dst[e]], 1.0f);
}

__global__ __launch_bounds__(256) void k_dis(float* deg) {
    int i = blockIdx.x * 256 + threadIdx.x;
    if (i < GN) deg[i] = rsqrtf(deg[i]);     // deg >= 1 always
}

__global__ __launch_bounds__(256) void k_norm(const int* __restrict__ src,
                                              const int* __restrict__ dst,
                                              const float* __restrict__ dis,
                                              float* __restrict__ norm) {
    int e = blockIdx.x * 256 + threadIdx.x;  // grid sized exactly E/256
    norm[e] = dis[src[e]] * dis[dst[e]];
}

// ---------------------------------------------------------------------------
// Dense transform: XW = (relu?)X @ W   via V_WMMA_F32_16X16X4_F32
// X: (GROWS x 64) row-major, W: (64 x 64) row-major.
// One wave computes one 16x16 output tile, K swept 64 in steps of 4.
// Block = 256 threads = 8 waves = 2 M-tiles x 4 N-tiles (32 rows / block).
// EXEC is all-ones everywhere (exact divisibility), as WMMA requires.
// RELU is a template parameter so no per-element cndmask is generated.
// ---------------------------------------------------------------------------
template <bool RELU>
__global__ __launch_bounds__(256)
void k_mm_wmma(const float* __restrict__ X, const float* __restrict__ W,
               float* __restrict__ XW) {
    const int wave = threadIdx.x >> 5;
    const int lane = threadIdx.x & 31;
    const int half = lane >> 4;                 // 0 | 1
    const int l16  = lane & 15;
    const int mBase = (blockIdx.x * 2 + (wave >> 2)) * 16;
    const int nBase = (wave & 3) * 16;

    const float* __restrict__ arow = X + (size_t)(mBase + l16) * GD;
    const float* __restrict__ bcol = W + nBase + l16;

    v8f acc = {0.f, 0.f, 0.f, 0.f, 0.f, 0.f, 0.f, 0.f};

    #pragma unroll
    for (int kk = 0; kk < GD; kk += 4) {
        const int k0 = kk + 2 * half;
        // A tile 16x4 fp32: lane holds A[l16][k0], A[l16][k0+1]  (contiguous)
        v2f a = *(const v2f*)(arow + k0);
        if (RELU) {
            a.x = relu1(a.x);
            a.y = relu1(a.y);
        }
        // B tile 4x16 fp32: lane holds B[k0][l16], B[k0+1][l16]
        v2f b;
        b.x = bcol[(size_t)(k0 + 0) * GD];
        b.y = bcol[(size_t)(k0 + 1) * GD];
        acc = __builtin_amdgcn_wmma_f32_16x16x4_f32(
                  /*neg_a=*/false, a, /*neg_b=*/false, b,
                  /*c_mod=*/(short)0, acc, /*reuse_a=*/false, /*reuse_b=*/false);
    }

    // D tile: VGPR v, lane -> D[v + 8*half][l16]
    float* __restrict__ orow = XW + (size_t)mBase * GD + nBase + l16;
    #pragma unroll
    for (int v = 0; v < 8; ++v) {
        orow[(size_t)(v + 8 * half) * GD] = acc[v];
    }
}

// ---------------------------------------------------------------------------
// Non-atomic init pass: out[b,n,:] = xw[b,n,:]*dis[n]^2 + bias (+ residual)
// (dis[n]^2 is the self-loop edge's norm). float4-vectorized.
// ---------------------------------------------------------------------------
__global__ __launch_bounds__(256)
void k_agg_init(const float* __restrict__ xw, const float* __restrict__ dis,
                const float* __restrict__ bias, const float* __restrict__ resid,
                float* __restrict__ out, int has_resid) {
    const size_t idx = (size_t)blockIdx.x * 256 + threadIdx.x;  // over B*N*D/4
    const int d4 = (int)(idx & 15);               // 16 float4 per row (D=64)
    const int n  = (int)((idx >> 4) % GN);
    const float s  = dis[n];
    const float ss = s * s;
    const float4 v  = ((const float4*)xw)[idx];
    const float4 bb = ((const float4*)bias)[d4];
    float4 o;
    o.x = v.x * ss + bb.x;
    o.y = v.y * ss + bb.y;
    o.z = v.z * ss + bb.z;
    o.w = v.w * ss + bb.w;
    if (has_resid) {
        const float4 r = ((const float4*)resid)[idx];
        o.x += r.x; o.y += r.y; o.z += r.z; o.w += r.w;
    }
    ((float4*)out)[idx] = o;
}

// ---------------------------------------------------------------------------
// Edge scatter: out[b,dst,:] += xw[b,src,:] * norm[e]
// One block per edge: 8 batches x 32 float2 slots = 256 threads.
// ---------------------------------------------------------------------------
__global__ __launch_bounds__(256)
void k_edge_scatter(const float* __restrict__ xw, const int* __restrict__ src,
                    const int* __restrict__ dst, const float* __restrict__ norm,
                    float* __restrict__ out) {
    const size_t idx = (size_t)blockIdx.x * 256 + threadIdx.x;
    const int d2 = (int)(idx & 31);          // float2 slot within row
    const int b  = (int)((idx >> 5) & 7);    // GB == 8
    const int e  = (int)(idx >> 8);          // one edge per block
    const int s = src[e];
    const int d = dst[e];
    const float nm = norm[e];
    const float2 m = *(const float2*)(xw + ((size_t)b * GN + s) * GD + d2 * 2);
    float* o = out + ((size_t)b * GN + d) * GD + d2 * 2;
    atomicAdd(o + 0, m.x * nm);
    atomicAdd(o + 1, m.y * nm);
}

// ---------------------------------------------------------------------------
// Copy init_h into the second output slot (tuple return), float4-vectorized.
// ---------------------------------------------------------------------------
__global__ __launch_bounds__(256)
void k_copy4(const float* __restrict__ in, float* __restrict__ out) {
    const size_t idx = (size_t)blockIdx.x * 256 + threadIdx.x;
    ((float4*)out)[idx] = ((const float4*)in)[idx];
}

// ---------------------------------------------------------------------------
extern "C" void kernel_launch(void* const* d_in, const int* in_sizes, int n_in,
                              void* d_out, int out_size, void* d_ws, size_t ws_size,
                              hipStream_t stream) {
    const float* init_h = (const float*)d_in[0];           // [B,N,D]
    const float* Wall   = (const float*)d_in[1];           // [L,D,D]
    const float* ball   = (const float*)d_in[2];           // [L,D]
    const int*   eidx   = (const int*)d_in[3];             // [2,E]
    const int*   src    = eidx;
    const int*   dst    = eidx + GE;

    float* out  = (float*)d_out;                           // [2, B,N,D] flat
    float* out2 = out + (size_t)GROWS * GD;

    // workspace layout (floats)
    float* ws   = (float*)d_ws;
    float* dis  = ws;                                      // N
    float* norm = ws + 10240;                              // E
    float* xw   = ws + 10240 + GE;                         // B*N*D
    float* hbuf = xw + (size_t)GROWS * GD;                 // B*N*D

    const int mm_blocks   = GROWS / 32;                    // 2500
    const int vec4_blocks = (GROWS * GD / 4) / 256;        // 5000
    const int edge_blocks = GE;                            // 1 edge / block
    const int n_blocks    = (GN + 255) / 256;
    const int e_blocks    = GE / 256;                      // exact

    // --- normalization: deg -> dis -> per-edge norm ---
    k_deg_init <<<n_blocks, 256, 0, stream>>>(dis);
    k_deg_count<<<e_blocks, 256, 0, stream>>>(dst, dis);
    k_dis      <<<n_blocks, 256, 0, stream>>>(dis);
    k_norm     <<<e_blocks, 256, 0, stream>>>(src, dst, dis, norm);

    // --- layer 0: h1 = relu(conv(init_h)) ---
    k_mm_wmma<false><<<mm_blocks, 256, 0, stream>>>(init_h, Wall + 0 * GD * GD, xw);
    k_agg_init      <<<vec4_blocks, 256, 0, stream>>>(xw, dis, ball + 0 * GD, nullptr, hbuf, 0);
    k_edge_scatter  <<<edge_blocks, 256, 0, stream>>>(xw, src, dst, norm, hbuf);

    // --- layer 1: h2 = relu(conv(h1))  (relu fused into A-load of mm) ---
    k_mm_wmma<true> <<<mm_blocks, 256, 0, stream>>>(hbuf, Wall + 1 * GD * GD, xw);
    k_agg_init      <<<vec4_blocks, 256, 0, stream>>>(xw, dis, ball + 1 * GD, nullptr, hbuf, 0);
    k_edge_scatter  <<<edge_blocks, 256, 0, stream>>>(xw, src, dst, norm, hbuf);

    // --- layer 2: out = conv(h2) + init_h ---
    k_mm_wmma<true> <<<mm_blocks, 256, 0, stream>>>(hbuf, Wall + 2 * GD * GD, xw);
    k_agg_init      <<<vec4_blocks, 256, 0, stream>>>(xw, dis, ball + 2 * GD, init_h, out, 1);
    k_edge_scatter  <<<edge_blocks, 256, 0, stream>>>(xw, src, dst, norm, out);

    // --- second tuple element: init_h passthrough ---
    k_copy4<<<vec4_blocks, 256, 0, stream>>>(init_h, out2);
}